// SparsemaxActivation_25271587570064
// MI455X (gfx1250) — compile-verified
//
#include <hip/hip_runtime.h>
#include <hip/hip_bf16.h>

// Masked sparsemax, rowwise over (8192, 4096) fp32.
//   masked = where(mask, scores, -inf);  out = sparsemax(masked, dim=-1)
// Algorithm: per row, bisection on tau in [zmax-1, zmax] (f(tau)=sum relu(z-tau)
// is piecewise-linear decreasing, f(zmax-1)>=1>=0=f(zmax)), 30 iters, then an
// exact polish tau=(S-1)/k over the recovered support {z > lo}.
//
// CDNA5 specifics: wave32, async global->LDS staging (global_load_async_to_lds_b128
// + s_wait_asynccnt, ASYNCcnt), 128-bit ds loads, shuffle-based wave32 reductions.
// WMMA is not applicable (no MMA structure in this op); the op is memory-bound:
// ~288 MiB of traffic -> ~12 us floor at 23.3 TB/s.

#define SROW    4096
#define THREADS 256      // 8 wave32 waves per row
#define EPT     16       // elements per thread (4 x float4)
#define NITER   30       // bisection iterations: interval 2^-30

// Block-wide sum over 256 threads; single barrier, LDS double-buffered by parity.
__device__ __forceinline__ float block_sum(float v, float* slots, int parity, int tid) {
#pragma unroll
  for (int off = 16; off >= 1; off >>= 1) v += __shfl_xor(v, off, 32);
  const int wave = tid >> 5, lane = tid & 31;
  float* buf = slots + (parity & 1) * 8;
  if (lane == 0) buf[wave] = v;
  __syncthreads();
  float s = 0.0f;
#pragma unroll
  for (int w = 0; w < 8; ++w) s += buf[w];
  return s;   // uniform across the block
}

__device__ __forceinline__ float block_max(float v, float* slots, int parity, int tid) {
#pragma unroll
  for (int off = 16; off >= 1; off >>= 1) v = fmaxf(v, __shfl_xor(v, off, 32));
  const int wave = tid >> 5, lane = tid & 31;
  float* buf = slots + (parity & 1) * 8;
  if (lane == 0) buf[wave] = v;
  __syncthreads();
  float m = -INFINITY;
#pragma unroll
  for (int w = 0; w < 8; ++w) m = fmaxf(m, buf[w]);
  return m;
}

__global__ __launch_bounds__(THREADS)
void sparsemax_rows_kernel(const float* __restrict__ scores,
                           const unsigned char* __restrict__ mask,
                           float* __restrict__ out) {
  __shared__ __align__(16) float         z_s[SROW];   // 16 KB staged row
  __shared__ __align__(16) unsigned char m_s[SROW];   // 4 KB staged mask bytes
  __shared__ float red_s[16];                         // 2 x 8-wave reduction buffers

  const int    tid   = threadIdx.x;
  const size_t rbase = (size_t)blockIdx.x * SROW;

  // ---- Stage row + mask into LDS via gfx1250 async global->LDS engine ----
  {
    const unsigned long long gz = (unsigned long long)(scores + rbase);
    const unsigned long long gm = (unsigned long long)(mask + rbase);
#pragma unroll
    for (int c = 0; c < 4; ++c) {
      const int fbase = c * 1024 + tid * 4;                       // float index
      unsigned lds_off = (unsigned)(unsigned long long)&z_s[fbase];
      unsigned long long ga = gz + (unsigned long long)fbase * 4ull;
      asm volatile("global_load_async_to_lds_b128 %0, %1, off"
                   :: "v"(lds_off), "v"(ga) : "memory");
    }
    {
      unsigned lds_off = (unsigned)(unsigned long long)&m_s[tid * 16];
      unsigned long long ga = gm + (unsigned long long)(tid * 16);
      asm volatile("global_load_async_to_lds_b128 %0, %1, off"
                   :: "v"(lds_off), "v"(ga) : "memory");
    }
    asm volatile("s_wait_asynccnt 0x0" ::: "memory");  // ASYNCcnt == 0
  }
  __syncthreads();

  // ---- Pull into registers (ds_load_b128), apply mask, local max ----
  float z[EPT];
  float lmax = -INFINITY;
#pragma unroll
  for (int c = 0; c < 4; ++c) {
    const int fbase = c * 1024 + tid * 4;
    const float4   v  = *reinterpret_cast<const float4*>(&z_s[fbase]);
    const unsigned mb = *reinterpret_cast<const unsigned*>(&m_s[fbase]);
    z[c * 4 + 0] = ((mb >>  0) & 0xffu) ? v.x : -INFINITY;
    z[c * 4 + 1] = ((mb >>  8) & 0xffu) ? v.y : -INFINITY;
    z[c * 4 + 2] = ((mb >> 16) & 0xffu) ? v.z : -INFINITY;
    z[c * 4 + 3] = ((mb >> 24) & 0xffu) ? v.w : -INFINITY;
#pragma unroll
    for (int j = 0; j < 4; ++j) lmax = fmaxf(lmax, z[c * 4 + j]);
  }

  int parity = 0;
  const float zmax = block_max(lmax, red_s, parity++, tid);

  // ---- Bisection on tau: invariant f(lo) >= 1 >= f(hi) ----
  float lo = zmax - 1.0f;
  float hi = zmax;
#pragma unroll 1
  for (int it = 0; it < NITER; ++it) {
    const float mid = 0.5f * (lo + hi);
    float s = 0.0f;
#pragma unroll
    for (int j = 0; j < EPT; ++j) s += fmaxf(z[j] - mid, 0.0f);
    s = block_sum(s, red_s, parity++, tid);
    if (s >= 1.0f) lo = mid; else hi = mid;   // uniform branch
  }

  // ---- Exact polish: support = {z > lo}, tau = (sum - 1) / k ----
  float sp = 0.0f, kp = 0.0f;
#pragma unroll
  for (int j = 0; j < EPT; ++j) {
    if (z[j] > lo) { sp += z[j]; kp += 1.0f; }
  }
  const float S_sum = block_sum(sp, red_s, parity++, tid);
  const float K_sum = block_sum(kp, red_s, parity++, tid);
  const float tau   = (S_sum - 1.0f) / fmaxf(K_sum, 1.0f);

  // ---- Write output (128-bit coalesced stores) ----
#pragma unroll
  for (int c = 0; c < 4; ++c) {
    const int fbase = c * 1024 + tid * 4;
    float4 o;
    o.x = fmaxf(z[c * 4 + 0] - tau, 0.0f);
    o.y = fmaxf(z[c * 4 + 1] - tau, 0.0f);
    o.z = fmaxf(z[c * 4 + 2] - tau, 0.0f);
    o.w = fmaxf(z[c * 4 + 3] - tau, 0.0f);
    *reinterpret_cast<float4*>(out + rbase + fbase) = o;
  }
}

extern "C" void kernel_launch(void* const* d_in, const int* in_sizes, int n_in,
                              void* d_out, int out_size, void* d_ws, size_t ws_size,
                              hipStream_t stream) {
  const float*         scores = (const float*)d_in[0];
  const unsigned char* mask   = (const unsigned char*)d_in[1];  // jnp.bool_: 1 byte/elem
  float*               out    = (float*)d_out;

  const int rows = in_sizes[0] / SROW;   // 8192
  sparsemax_rows_kernel<<<rows, THREADS, 0, stream>>>(scores, mask, out);
}